// SupernodePooling_38062000177493
// MI455X (gfx1250) — compile-verified
//
#include <hip/hip_runtime.h>
#include <math.h>

typedef __attribute__((ext_vector_type(2))) float v2f;
typedef __attribute__((ext_vector_type(8))) float v8f;
typedef __attribute__((ext_vector_type(4))) float v4f;
typedef __attribute__((ext_vector_type(4))) unsigned int u32x4;
typedef __attribute__((ext_vector_type(4))) int i32x4;
typedef __attribute__((ext_vector_type(8))) int i32x8;

constexpr int N_PTS   = 65536;
constexpr int S_SUP   = 2048;
constexpr int IN_DIM  = 32;
constexpr int HID     = 256;
constexpr int KDEG    = 32;
constexpr float RADIUS2 = 0.005f * 0.005f;
constexpr int CAND_CAP = 4096;

__device__ inline float gelu_tanh(float v) {
    float u = 0.7978845608028654f * (v + 0.044715f * v * v * v);
    return 0.5f * v * (1.0f + tanhf(u));
}

// ---------------------------------------------------------------------------
// Tensor Data Mover: 2-D tile load global -> LDS.
// D# packed per CDNA5 ISA ch.8 (group0: count/lds/global/type; group1:
// data_size=4B, tensor dims, tile dims, dim0 stride; groups 2/3 zero = 2-D).
// ---------------------------------------------------------------------------
__device__ inline void tdm_load_2d(unsigned lds_addr, const void* gptr,
                                   unsigned tensor_d0, unsigned tensor_d1,
                                   unsigned stride0,
                                   unsigned tile_d0, unsigned tile_d1) {
    unsigned long long ga = (unsigned long long)(uintptr_t)gptr;
    u32x4 g0;
    g0.x = 1u;                                        // count=1 (valid user D#)
    g0.y = lds_addr;                                  // LDS byte address
    g0.z = (unsigned)(ga & 0xFFFFFFFFu);              // global_addr[31:0]
    g0.w = (unsigned)((ga >> 32) & 0x01FFFFFFu)       // global_addr[56:32]
         | 0x80000000u;                               // type=2 ("image")
    i32x8 g1;
    g1[0] = 0x00020000;                               // data_size=2 -> 4 bytes
    g1[1] = (int)((tensor_d0 & 0xFFFFu) << 16);                       // dim0[15:0]
    g1[2] = (int)(((tensor_d0 >> 16) & 0xFFFFu) |
                  ((tensor_d1 & 0xFFFFu) << 16));                     // dim0[31:16] | dim1[15:0]
    g1[3] = (int)(((tensor_d1 >> 16) & 0xFFFFu) |
                  ((tile_d0 & 0xFFFFu) << 16));                       // dim1[31:16] | tile0
    g1[4] = (int)(tile_d1 & 0xFFFFu);                                 // tile1, tile2=0
    g1[5] = (int)stride0;                                             // dim0_stride[31:0]
    g1[6] = 0;
    g1[7] = 0;
    i32x4 z4 = {0, 0, 0, 0};
#if defined(__clang_major__) && __clang_major__ >= 23
    i32x8 z8 = {0, 0, 0, 0, 0, 0, 0, 0};
    __builtin_amdgcn_tensor_load_to_lds(g0, g1, z4, z4, z8, 0);
#else
    __builtin_amdgcn_tensor_load_to_lds(g0, g1, z4, z4, 0);
#endif
}

// ---------------------------------------------------------------------------
// Kernel 1: x = feat @ proj_W + proj_b + sincos_embed(pos)
// proj_W staged in LDS via TDM; 4 waves, one 16-row M-tile each.
// ---------------------------------------------------------------------------
__global__ void sp_embed_kernel(const float* __restrict__ feat,
                                const float* __restrict__ pos,
                                const float* __restrict__ W,
                                const float* __restrict__ bias,
                                float* __restrict__ x) {
    __shared__ float s_w[IN_DIM * HID];               // 32 KB weight stage

    const int lane = threadIdx.x & 31;
    const int wave = threadIdx.x >> 5;

    if (wave == 0) {
        tdm_load_2d((unsigned)(uintptr_t)s_w, W,
                    /*tensor_d0=*/HID, /*tensor_d1=*/IN_DIM,
                    /*stride0=*/HID, /*tile_d0=*/HID, /*tile_d1=*/IN_DIM);
    }
    __builtin_amdgcn_s_wait_tensorcnt(0);
    __syncthreads();

    const int rowBase = (blockIdx.x * 4 + wave) * 16;
    const int m    = lane & 15;            // A-row / B-col within tile
    const int koff = (lane < 16) ? 0 : 2;  // K selector per lane-half

    // Preload A for all 8 K-steps (K=32, 4 per WMMA).
    v2f a[8];
    const float* arow = feat + (size_t)(rowBase + m) * IN_DIM;
#pragma unroll
    for (int kk = 0; kk < 8; ++kk)
        a[kk] = *(const v2f*)(arow + kk * 4 + koff);

    const float inv_scale = -logf(10000.0f) / 127.0f;  // half-1 = 127

    for (int n0 = 0; n0 < HID; n0 += 16) {
        v8f acc = {};
        const float* bcol = s_w + n0 + m;
#pragma unroll
        for (int kk = 0; kk < 8; ++kk) {
            v2f bm;
            bm.x = bcol[(kk * 4 + koff) * HID];
            bm.y = bcol[(kk * 4 + koff + 1) * HID];
            acc = __builtin_amdgcn_wmma_f32_16x16x4_f32(
                false, a[kk], false, bm, (short)0, acc, false, false);
        }
        const int col = n0 + m;
#pragma unroll
        for (int v = 0; v < 8; ++v) {
            int row = rowBase + v + ((lane < 16) ? 0 : 8);
            float p = pos[row];
            float pe;
            if (col < 128) pe = __sinf(p * __expf((float)col * inv_scale));
            else           pe = __cosf(p * __expf((float)(col - 128) * inv_scale));
            x[(size_t)row * HID + col] = acc[v] + bias[col] + pe;
        }
    }
}

// ---------------------------------------------------------------------------
// Kernel 2: per-supernode neighbor selection.
// top-32-nearest masked to radius == the min(32,cnt) nearest in-radius points.
// ---------------------------------------------------------------------------
__global__ void sp_select_kernel(const float* __restrict__ pos,
                                 const int* __restrict__ sup_idx,
                                 int* __restrict__ nbr) {
    __shared__ float cd2[CAND_CAP];
    __shared__ int   cidx[CAND_CAP];
    __shared__ float rd[256];
    __shared__ int   ri[256];
    __shared__ int   scnt;

    const int s = blockIdx.x;
    const int tid = threadIdx.x;
    const float ps = pos[sup_idx[s]];

    if (tid == 0) scnt = 0;
    __syncthreads();

    for (int i = tid; i < N_PTS; i += 256) {
        float d = pos[i] - ps;
        float d2 = d * d;
        if (d2 <= RADIUS2) {
            int slot = atomicAdd(&scnt, 1);
            if (slot < CAND_CAP) { cd2[slot] = d2; cidx[slot] = i; }
        }
    }
    __syncthreads();
    const int m = (scnt < CAND_CAP) ? scnt : CAND_CAP;

    for (int r = 0; r < KDEG; ++r) {
        float bd = 3.0e38f;
        int   bi = 0;
        for (int i = tid; i < m; i += 256) {
            float d = cd2[i];
            if (d < bd) { bd = d; bi = i; }
        }
        rd[tid] = bd; ri[tid] = bi;
        __syncthreads();
        for (int off = 128; off > 0; off >>= 1) {
            if (tid < off && rd[tid + off] < rd[tid]) {
                rd[tid] = rd[tid + off]; ri[tid] = ri[tid + off];
            }
            __syncthreads();
        }
        if (tid == 0) {
            if (r < m) { nbr[s * KDEG + r] = cidx[ri[0]]; cd2[ri[0]] = 3.0e38f; }
            else       { nbr[s * KDEG + r] = -1; }
        }
        __syncthreads();
    }
}

// ---------------------------------------------------------------------------
// Kernel 3: per-supernode edge MLP + masked mean pooling.
// 256 threads (8 waves). Weight panels (64 cols) DMA'd to LDS by the TDM;
// each wave computes one 16x16 WMMA tile per N-group from LDS only.
// LDS: m_in[32][512] (64K, reused for m), h[32][256] (32K), panel 512x64 (128K).
// ---------------------------------------------------------------------------
__global__ void sp_mlp_pool_kernel(const float* __restrict__ x,
                                   const int* __restrict__ sup_idx,
                                   const int* __restrict__ nbr,
                                   const float* __restrict__ W1,
                                   const float* __restrict__ b1,
                                   const float* __restrict__ W2,
                                   const float* __restrict__ b2,
                                   float* __restrict__ out) {
    extern __shared__ char smem[];
    float* s_min = (float*)smem;                 // [32][512]; later m[32][256]
    float* s_h   = s_min + 32 * 512;             // [32][256]
    float* s_w   = s_h + 32 * 256;               // [512][64] weight panel
    int*   s_nbr = (int*)(s_w + 512 * 64);       // [32]

    const int s = blockIdx.x;
    const int tid = threadIdx.x;
    const int lane = tid & 31;
    const int wave = tid >> 5;

    if (tid < KDEG) s_nbr[tid] = nbr[s * KDEG + tid];
    __syncthreads();

    // Gather m_in = [x_src | x_dst] with 128-bit vectors.
    const float* xdst = x + (size_t)sup_idx[s] * HID;
    for (int t = tid; t < KDEG * (HID / 4); t += 256) {
        int e = t >> 6, c4 = (t & 63) * 4;
        int idx = s_nbr[e];
        v4f src = (idx >= 0) ? *(const v4f*)(x + (size_t)idx * HID + c4)
                             : (v4f){0.f, 0.f, 0.f, 0.f};
        *(v4f*)(s_min + e * 512 + c4)       = src;
        *(v4f*)(s_min + e * 512 + 256 + c4) = *(const v4f*)(xdst + c4);
    }

    const int m16  = lane & 15;
    const int koff = (lane < 16) ? 0 : 2;
    const int mt   = wave & 1;                   // M-tile (edges 0-15 / 16-31)
    const int ntl  = wave >> 1;                  // local N-tile within group
    const int colL = ntl * 16 + m16;             // column within 64-col panel

    // ---- Layer 1: h = gelu(m_in @ W1 + b1), K = 512, 4 N-groups of 64 ----
    for (int g = 0; g < 4; ++g) {
        __syncthreads();                          // panel free & m_in ready
        if (wave == 0) {
            tdm_load_2d((unsigned)(uintptr_t)s_w, W1 + g * 64,
                        /*tensor_d0=*/HID, /*tensor_d1=*/2 * HID,
                        /*stride0=*/HID, /*tile_d0=*/64, /*tile_d1=*/2 * HID);
        }
        __builtin_amdgcn_s_wait_tensorcnt(0);
        __syncthreads();

        const float* arow = s_min + (mt * 16 + m16) * 512;
        const float* bcol = s_w + colL;
        v8f acc = {};
#pragma unroll 4
        for (int k0 = 0; k0 < 512; k0 += 4) {
            v2f a = *(const v2f*)(arow + k0 + koff);
            v2f b;
            b.x = bcol[(k0 + koff) * 64];
            b.y = bcol[(k0 + koff + 1) * 64];
            acc = __builtin_amdgcn_wmma_f32_16x16x4_f32(
                false, a, false, b, (short)0, acc, false, false);
        }
        int col = g * 64 + colL;
#pragma unroll
        for (int v = 0; v < 8; ++v) {
            int row = mt * 16 + v + ((lane < 16) ? 0 : 8);
            s_h[row * HID + col] = gelu_tanh(acc[v] + b1[col]);
        }
    }
    __syncthreads();

    // ---- Layer 2: m = h @ W2 + b2, K = 256; m overwrites m_in region ----
    float* s_m = s_min;
    for (int g = 0; g < 4; ++g) {
        __syncthreads();
        if (wave == 0) {
            tdm_load_2d((unsigned)(uintptr_t)s_w, W2 + g * 64,
                        /*tensor_d0=*/HID, /*tensor_d1=*/HID,
                        /*stride0=*/HID, /*tile_d0=*/64, /*tile_d1=*/HID);
        }
        __builtin_amdgcn_s_wait_tensorcnt(0);
        __syncthreads();

        const float* arow = s_h + (mt * 16 + m16) * HID;
        const float* bcol = s_w + colL;
        v8f acc = {};
#pragma unroll 4
        for (int k0 = 0; k0 < 256; k0 += 4) {
            v2f a = *(const v2f*)(arow + k0 + koff);
            v2f b;
            b.x = bcol[(k0 + koff) * 64];
            b.y = bcol[(k0 + koff + 1) * 64];
            acc = __builtin_amdgcn_wmma_f32_16x16x4_f32(
                false, a, false, b, (short)0, acc, false, false);
        }
        int col = g * 64 + colL;
#pragma unroll
        for (int v = 0; v < 8; ++v) {
            int row = mt * 16 + v + ((lane < 16) ? 0 : 8);
            s_m[row * HID + col] = acc[v] + b2[col];
        }
    }
    __syncthreads();

    // ---- Masked mean over valid edges: each thread owns one column ----
    float sum = 0.0f;
    int cnt = 0;
#pragma unroll
    for (int e = 0; e < KDEG; ++e) {
        if (s_nbr[e] >= 0) { sum += s_m[e * HID + tid]; ++cnt; }
    }
    out[(size_t)s * HID + tid] = sum / (float)((cnt > 0) ? cnt : 1);
}

// ---------------------------------------------------------------------------
extern "C" void kernel_launch(void* const* d_in, const int* in_sizes, int n_in,
                              void* d_out, int out_size, void* d_ws, size_t ws_size,
                              hipStream_t stream) {
    const float* feat  = (const float*)d_in[0];
    const float* pos   = (const float*)d_in[1];
    const int*   sup   = (const int*)d_in[2];
    // d_in[3] = batch_idx (single batch) — unused
    const float* projW = (const float*)d_in[4];
    const float* projB = (const float*)d_in[5];
    const float* W1    = (const float*)d_in[6];
    const float* b1    = (const float*)d_in[7];
    const float* W2    = (const float*)d_in[8];
    const float* b2    = (const float*)d_in[9];
    float* out = (float*)d_out;

    float* x   = (float*)d_ws;                                   // 64 MB
    int*   nbr = (int*)((char*)d_ws + (size_t)N_PTS * HID * sizeof(float));

    sp_embed_kernel<<<N_PTS / 64, 128, 0, stream>>>(feat, pos, projW, projB, x);
    sp_select_kernel<<<S_SUP, 256, 0, stream>>>(pos, sup, nbr);

    size_t lds = (32 * 512 + 32 * 256 + 512 * 64) * sizeof(float)
               + KDEG * sizeof(int);
    sp_mlp_pool_kernel<<<S_SUP, 256, lds, stream>>>(x, sup, nbr, W1, b1, W2, b2, out);
}